// cls_model_40664750359206
// MI455X (gfx1250) — compile-verified
//
#include <hip/hip_runtime.h>
#include <hip/hip_bf16.h>

// Problem constants (from setup_inputs: B=8, N=2048, Fin=6, Fout=128, K=6, k=40)
#define BB    8
#define NN    2048
#define FIN   6
#define FOUT  128
#define KCH   6
#define KNN   40
#define KF    (KCH * FIN)      // 36 stacked Chebyshev features
#define NROWS (BB * NN)        // 16384 graph nodes total
#define NCLS  40
#define KSLAB 64               // K-depth staged in LDS per Gram chunk

typedef __attribute__((ext_vector_type(2))) float v2f;
typedef __attribute__((ext_vector_type(8))) float v8f;

// CDNA5 async global->LDS copy (ASYNCcnt-tracked). VDST = LDS byte offset,
// VADDR = 64-bit global address, saddr = off (GV mode).
__device__ __forceinline__ void async_load_b128(unsigned lds_off, const void* gaddr)
{
    asm volatile("global_load_async_to_lds_b128 %0, %1, off"
                 :: "v"(lds_off), "v"(gaddr) : "memory");
}
__device__ __forceinline__ void wait_asynccnt0()
{
    asm volatile("s_wait_asynccnt 0x0" ::: "memory");
}

// ---------------------------------------------------------------------------
// 1) Pairwise sq-distance row + top-40 selection.
//    exp(-d2) is monotone-decreasing in d2, so top-k of A == bottom-k of d2.
//    Selection: per-round wave32 shfl argmin (no barriers) + 8-entry LDS merge.
//    Also emits dis = 1/sqrt(degree) since degree = sum of kept values.
// ---------------------------------------------------------------------------
__global__ __launch_bounds__(256) void topk_kernel(
    const float* __restrict__ x, float* __restrict__ wval,
    int* __restrict__ widx, float* __restrict__ dis)
{
    __shared__ float d2[NN];
    __shared__ float rv[8];
    __shared__ int   ri[8];

    const int row = blockIdx.x;            // b*NN + i
    const int b = row / NN, i = row % NN;
    const float* xb = x + (size_t)b * NN * FIN;

    float xi[FIN];
#pragma unroll
    for (int f = 0; f < FIN; ++f) xi[f] = xb[(size_t)i * FIN + f];

    for (int j = threadIdx.x; j < NN; j += 256) {
        float s = 0.f;
#pragma unroll
        for (int f = 0; f < FIN; ++f) {
            float dd = xi[f] - xb[(size_t)j * FIN + f];
            s += dd * dd;
        }
        d2[j] = s;
    }
    __syncthreads();

    const int wave = threadIdx.x >> 5;
    const int lane = threadIdx.x & 31;
    float degree = 0.f;                    // only thread 0 accumulates

    for (int t = 0; t < KNN; ++t) {
        float best = 3.402823466e+38f;
        int   bi   = NN;
        for (int j = threadIdx.x; j < NN; j += 256) {
            float v = d2[j];
            if (v < best) { best = v; bi = j; }   // keeps lowest j on ties
        }
        // wave32 argmin reduction, tie -> lowest index (matches jax top_k)
#pragma unroll
        for (int d = 16; d > 0; d >>= 1) {
            float ov = __shfl_down(best, d, 32);
            int   oi = __shfl_down(bi, d, 32);
            if (ov < best || (ov == best && oi < bi)) { best = ov; bi = oi; }
        }
        if (lane == 0) { rv[wave] = best; ri[wave] = bi; }
        __syncthreads();
        if (threadIdx.x == 0) {
            float mv = rv[0]; int mi = ri[0];
#pragma unroll
            for (int u = 1; u < 8; ++u) {
                float ov = rv[u]; int oi = ri[u];
                if (ov < mv || (ov == mv && oi < mi)) { mv = ov; mi = oi; }
            }
            float a = expf(-mv);
            wval[(size_t)row * KNN + t] = a;
            widx[(size_t)row * KNN + t] = mi;
            degree += a;
            d2[mi] = 3.402823466e+38f;     // exclude from later rounds
        }
        __syncthreads();
    }
    if (threadIdx.x == 0) dis[row] = 1.0f / sqrtf(degree);
}

// ---------------------------------------------------------------------------
// 2) In-place: Wval -> Lval = -W * dis_i * dis_j (diagonal "I" handled in SpMV)
// ---------------------------------------------------------------------------
__global__ void lap_kernel(float* __restrict__ wval, const int* __restrict__ widx,
                           const float* __restrict__ dis)
{
    int g = blockIdx.x * blockDim.x + threadIdx.x;     // over NROWS*KNN
    if (g >= NROWS * KNN) return;
    int row = g / KNN;
    int b   = row / NN;
    wval[g] = -wval[g] * dis[row] * dis[b * NN + widx[g]];
}

// ---------------------------------------------------------------------------
// 3) Pack x into Chebyshev slab 0 of Xc[NROWS][KF]
// ---------------------------------------------------------------------------
__global__ void pack_x0(const float* __restrict__ x, float* __restrict__ Xc)
{
    int g = blockIdx.x * blockDim.x + threadIdx.x;     // over NROWS*FIN
    if (g >= NROWS * FIN) return;
    int row = g / FIN, f = g % FIN;
    Xc[(size_t)row * KF + f] = x[g];
}

// ---------------------------------------------------------------------------
// 4) One Chebyshev step: y = L @ x_{k-1}; x_k = (k==1) ? y : 2y - x_{k-2}
// ---------------------------------------------------------------------------
__global__ void cheb_step(const float* __restrict__ lval, const int* __restrict__ lidx,
                          float* __restrict__ Xc, int kstep)
{
    int g = blockIdx.x * blockDim.x + threadIdx.x;     // over NROWS*FIN
    if (g >= NROWS * FIN) return;
    int f = g % FIN, row = g / FIN, b = row / NN;

    const size_t base = (size_t)row * KF;
    const int   prev  = (kstep - 1) * FIN + f;
    const float* rval = lval + (size_t)row * KNN;
    const int*   ridx = lidx + (size_t)row * KNN;
    const int    boff = b * NN;

    float acc = Xc[base + prev];                       // diagonal of L (the I term)
    for (int t = 0; t < KNN; ++t)
        acc += rval[t] * Xc[(size_t)(boff + ridx[t]) * KF + prev];

    float res = (kstep == 1) ? acc : (2.0f * acc - Xc[base + (kstep - 2) * FIN + f]);
    Xc[base + kstep * FIN + f] = res;
}

// ---------------------------------------------------------------------------
// 5) out = relu(Xc[16384,36] @ Wflat[36,128] + bias) via V_WMMA_F32_16X16X4_F32
//    A 16x4 f32 layout: lane L -> m = L&15, a[j] = A[m][2*(L>>4)+j]
//    B 4x16 mirrored;  C/D: lane L -> n = L&15, VGPR v -> m = v + 8*(L>>4)
// ---------------------------------------------------------------------------
__global__ __launch_bounds__(32) void conv_wmma(
    const float* __restrict__ Xc, const float* __restrict__ w,
    const float* __restrict__ bias, float* __restrict__ out)
{
    const int tile = blockIdx.x;        // 1024 m-tiles * 8 n-tiles
    const int tn   = tile & 7;
    const int tm   = tile >> 3;
    const int lane = threadIdx.x;
    const int half = lane >> 4;
    const int lan  = lane & 15;
    const int rowA = tm * 16 + lan;
    const int colB = tn * 16 + lan;

    v8f c = {};
#pragma unroll
    for (int kk = 0; kk < KF / 4; ++kk) {
        const int k0 = kk * 4 + half * 2;
        v2f a, bf;
        a.x  = Xc[(size_t)rowA * KF + k0];
        a.y  = Xc[(size_t)rowA * KF + k0 + 1];
        bf.x = w[(size_t)k0 * FOUT + colB];
        bf.y = w[(size_t)(k0 + 1) * FOUT + colB];
        c = __builtin_amdgcn_wmma_f32_16x16x4_f32(false, a, false, bf,
                                                  (short)0, c, false, false);
    }
    const float bb = bias[colB];
#pragma unroll
    for (int v = 0; v < 8; ++v) {
        const int m = v + 8 * half;
        float r = c[v] + bb;
        r = r > 0.f ? r : 0.f;
        out[(size_t)(tm * 16 + m) * FOUT + colB] = r;
    }
}

// ---------------------------------------------------------------------------
// 6) M = L @ out  (sparse rows, 128-wide, coalesced over the feature dim)
// ---------------------------------------------------------------------------
__global__ __launch_bounds__(128) void spmm_kernel(
    const float* __restrict__ lval, const int* __restrict__ lidx,
    const float* __restrict__ out, float* __restrict__ Mbuf)
{
    const int row = blockIdx.x;         // b*NN + i
    const int o   = threadIdx.x;        // 0..127
    const int b   = row / NN;
    __shared__ float sv[KNN];
    __shared__ int   si[KNN];
    if (threadIdx.x < KNN) {
        sv[threadIdx.x] = lval[(size_t)row * KNN + threadIdx.x];
        si[threadIdx.x] = lidx[(size_t)row * KNN + threadIdx.x];
    }
    __syncthreads();
    float acc = out[(size_t)row * FOUT + o];           // diagonal (I) term
    const int boff = b * NN;
    for (int t = 0; t < KNN; ++t)
        acc += sv[t] * out[(size_t)(boff + si[t]) * FOUT + o];
    Mbuf[(size_t)row * FOUT + o] = acc;
}

// ---------------------------------------------------------------------------
// 7) Gram: G[b] = out_bᵀ @ M_b, accumulate ||G||_F² partial per block.
//    One 8-wave block per (batch, f-tile); K staged in 64-deep LDS slabs via
//    async global->LDS b128 copies; all 8 waves (tg = waveID) share the B slab.
// ---------------------------------------------------------------------------
__global__ __launch_bounds__(256) void gram_wmma(
    const float* __restrict__ out, const float* __restrict__ Mbuf,
    float* __restrict__ partial)
{
    __shared__ float Asl[KSLAB * 16];      // out[k][tf*16 .. tf*16+15]
    __shared__ float Bsl[KSLAB * FOUT];    // M[k][0..127]
    __shared__ float red[256];

    const int blk = blockIdx.x;            // BB * 8 = 64 blocks
    const int b   = blk >> 3;
    const int tf  = blk & 7;
    const int tid = threadIdx.x;
    const int wv  = tid >> 5;              // tg = wave id (0..7)
    const int lane = tid & 31;
    const int half = lane >> 4;
    const int lan  = lane & 15;

    const float* ob = out  + (size_t)b * NN * FOUT;
    const float* mb = Mbuf + (size_t)b * NN * FOUT;

    const unsigned aBase = (unsigned)(uintptr_t)(void*)Asl;
    const unsigned bBase = (unsigned)(uintptr_t)(void*)Bsl;

    // Per-thread fixed (row, segment) assignments for the async copies
    const int ar = tid >> 2, aq = tid & 3;     // A: 64 rows x 4 x b128
    v8f c = {};

    for (int kc = 0; kc < NN; kc += KSLAB) {
        // A slab: 64 x 16 floats
        async_load_b128(aBase + (unsigned)(ar * 16 + aq * 4) * 4,
                        ob + (size_t)(kc + ar) * FOUT + tf * 16 + aq * 4);
        // B slab: 64 x 128 floats = 2048 b128 segments, 8 per thread
#pragma unroll
        for (int i = 0; i < 8; ++i) {
            const int e  = tid + i * 256;
            const int br = e >> 5, bq = e & 31;
            async_load_b128(bBase + (unsigned)(br * FOUT + bq * 4) * 4,
                            mb + (size_t)(kc + br) * FOUT + bq * 4);
        }
        wait_asynccnt0();
        __syncthreads();

#pragma unroll
        for (int kk = 0; kk < KSLAB / 4; ++kk) {
            const int k0 = kk * 4 + half * 2;
            v2f a, bf;
            a.x  = Asl[k0 * 16 + lan];
            a.y  = Asl[(k0 + 1) * 16 + lan];
            bf.x = Bsl[k0 * FOUT + wv * 16 + lan];
            bf.y = Bsl[(k0 + 1) * FOUT + wv * 16 + lan];
            c = __builtin_amdgcn_wmma_f32_16x16x4_f32(false, a, false, bf,
                                                      (short)0, c, false, false);
        }
        __syncthreads();                   // slab reuse next iteration
    }

    float s = 0.f;
#pragma unroll
    for (int v = 0; v < 8; ++v) s += c[v] * c[v];
    red[tid] = s;
    __syncthreads();
    for (int st = 128; st > 0; st >>= 1) {
        if (tid < st) red[tid] += red[tid + st];
        __syncthreads();
    }
    if (tid == 0) partial[blk] = red[0];
}

// 8) Deterministic reduction of 64 block partials -> reg at d_out[320]
__global__ __launch_bounds__(64) void reduce_reg(const float* __restrict__ partial,
                                                 float* __restrict__ outp)
{
    __shared__ float red[64];
    red[threadIdx.x] = partial[threadIdx.x];
    __syncthreads();
    for (int st = 32; st > 0; st >>= 1) {
        if (threadIdx.x < (unsigned)st) red[threadIdx.x] += red[threadIdx.x + st];
        __syncthreads();
    }
    if (threadIdx.x == 0) outp[BB * NCLS] = red[0];
}

// 9) Global max pool over N (coalesced over features)
__global__ __launch_bounds__(128) void pool_kernel(const float* __restrict__ out,
                                                   float* __restrict__ pooled)
{
    const int b = blockIdx.x, o = threadIdx.x;
    const float* p = out + (size_t)b * NN * FOUT + o;
    float m = p[0];
    for (int n = 1; n < NN; ++n) {
        float v = p[(size_t)n * FOUT];
        m = v > m ? v : m;
    }
    pooled[b * FOUT + o] = m;
}

// 10) logits = pooled @ fc_w.T + fc_b -> d_out[0..319]
__global__ __launch_bounds__(64) void fc_kernel(
    const float* __restrict__ pooled, const float* __restrict__ fw,
    const float* __restrict__ fb, float* __restrict__ outp)
{
    const int b = blockIdx.x, c = threadIdx.x;
    if (c < NCLS) {
        float acc = fb[c];
        const float* p = pooled + b * FOUT;
        for (int o = 0; o < FOUT; ++o) acc += p[o] * fw[(size_t)c * FOUT + o];
        outp[b * NCLS + c] = acc;
    }
}

extern "C" void kernel_launch(void* const* d_in, const int* in_sizes, int n_in,
                              void* d_out, int out_size, void* d_ws, size_t ws_size,
                              hipStream_t stream)
{
    (void)in_sizes; (void)n_in; (void)out_size; (void)ws_size;
    const float* x       = (const float*)d_in[0];   // [8,2048,6]
    const float* conv1_w = (const float*)d_in[1];   // [6,6,128] -> flat [36,128]
    const float* conv1_b = (const float*)d_in[2];   // [128]
    const float* fc_w    = (const float*)d_in[3];   // [40,128]
    const float* fc_b    = (const float*)d_in[4];   // [40]
    float* outp = (float*)d_out;                    // 320 logits + 1 reg

    // Workspace carve-up (~24.2 MB total)
    char* ws = (char*)d_ws;
    size_t off = 0;
    auto carve = [&](size_t bytes) -> void* {
        void* p = ws + off;
        off += (bytes + 255) & ~(size_t)255;
        return p;
    };
    float* wval    = (float*)carve((size_t)NROWS * KNN * 4);   // becomes Lval in place
    int*   widx    = (int*)  carve((size_t)NROWS * KNN * 4);
    float* dis     = (float*)carve((size_t)NROWS * 4);
    float* Xc      = (float*)carve((size_t)NROWS * KF * 4);
    float* outbuf  = (float*)carve((size_t)NROWS * FOUT * 4);
    float* Mbuf    = (float*)carve((size_t)NROWS * FOUT * 4);
    float* pooled  = (float*)carve((size_t)BB * FOUT * 4);
    float* partial = (float*)carve(64 * 4);

    // 1) distances + top-40 + dis
    topk_kernel<<<NROWS, 256, 0, stream>>>(x, wval, widx, dis);
    // 2) normalized Laplacian off-diagonal values (in place)
    lap_kernel<<<(NROWS * KNN + 255) / 256, 256, 0, stream>>>(wval, widx, dis);
    // 3-4) Chebyshev feature stack
    pack_x0<<<(NROWS * FIN + 255) / 256, 256, 0, stream>>>(x, Xc);
    for (int k = 1; k < KCH; ++k)
        cheb_step<<<(NROWS * FIN + 255) / 256, 256, 0, stream>>>(wval, widx, Xc, k);
    // 5) conv GEMM (WMMA) + bias + relu
    conv_wmma<<<(NROWS / 16) * (FOUT / 16), 32, 0, stream>>>(Xc, conv1_w, conv1_b, outbuf);
    // 6) M = L @ out
    spmm_kernel<<<NROWS, 128, 0, stream>>>(wval, widx, outbuf, Mbuf);
    // 7-8) reg = ||outᵀ L out||_F²  (async-LDS staged WMMA)
    gram_wmma<<<BB * 8, 256, 0, stream>>>(outbuf, Mbuf, partial);
    reduce_reg<<<1, 64, 0, stream>>>(partial, outp);
    // 9-10) max pool + classifier
    pool_kernel<<<BB, 128, 0, stream>>>(outbuf, pooled);
    fc_kernel<<<BB, 64, 0, stream>>>(pooled, fc_w, fc_b, outp);
}